// A_76278619177037
// MI455X (gfx1250) — compile-verified
//
#include <hip/hip_runtime.h>

// out[b, :] = z[b, :] + aT[idx[b], :] * scale[b]
//   z:   (B, 128) f32      B = 1048576
//   a:   (128, 512) f32    -> transposed once into d_ws as aT (512, 128)
//   idx: (B,) int          (harness passes integer inputs as int32)
//   scale: (B,) f32
// Memory-bound: ~1.04 GB HBM traffic -> ~45 us at 23.3 TB/s.

#define B_ROWS   1048576
#define ZDIM     128
#define KDIM     512
#define TBLK     1024      // 32 waves per workgroup (wave32)
#define NBLK     1024      // grid-stride persistent-ish blocks

typedef float v4f __attribute__((ext_vector_type(4)));
typedef int   v4i __attribute__((ext_vector_type(4)));

// Pointer types matching the gfx1250 async-to-LDS builtin signature:
//   (v4i addrspace(1)* src, v4i addrspace(3)* dst, imm offset, imm cpol)
typedef __attribute__((address_space(1))) v4i* g_v4i_ptr;
typedef __attribute__((address_space(3))) v4i* l_v4i_ptr;

// ---------- CDNA5 async global->LDS staging (guarded) ----------
#if defined(__has_builtin)
#  if __has_builtin(__builtin_amdgcn_global_load_async_to_lds_b128) && \
      __has_builtin(__builtin_amdgcn_s_wait_asynccnt)
#    define USE_ASYNC_LDS 1
#  endif
#endif

// ---------------------------------------------------------------
// Hot kernel first (so the disasm snippet shows it):
// stage aT (256 KB) into LDS via async b128, then stream rows.
// One wave per row: 32 lanes x float4 == 128 floats.
// ---------------------------------------------------------------
__global__ __launch_bounds__(TBLK)
void A_fused_kernel(const float* __restrict__ z,
                    const float* __restrict__ aT,
                    const int*   __restrict__ idx,
                    const float* __restrict__ scale,
                    float*       __restrict__ out) {
    __shared__ float ldsA[KDIM * ZDIM];  // 262144 B -- CDNA5 WGP has 320 KB LDS

    const int tid = threadIdx.x;

    // ---- stage aT -> LDS, contiguous b128 both sides (bank-conflict free) ----
#if USE_ASYNC_LDS
    for (int i = tid * 4; i < KDIM * ZDIM; i += TBLK * 4) {
        __builtin_amdgcn_global_load_async_to_lds_b128(
            (g_v4i_ptr)(aT + i),          // global src (AS1, v4i*)
            (l_v4i_ptr)(&ldsA[i]),        // LDS dst    (AS3, v4i*)
            /*offset=*/0, /*cpol=*/0);
    }
    __builtin_amdgcn_s_wait_asynccnt(0);
#else
    {
        const v4f* __restrict__ s4 = (const v4f*)aT;
        v4f* d4 = (v4f*)ldsA;
        for (int i = tid; i < (KDIM * ZDIM) / 4; i += TBLK) d4[i] = s4[i];
    }
#endif
    __syncthreads();

    // ---- stream rows: one wave32 per row, 4 rows in flight per wave ----
    const int  lane = tid & 31;
    const int  wave = tid >> 5;
    const long gw   = (long)blockIdx.x * (TBLK / 32) + wave;
    const long nw   = (long)gridDim.x * (TBLK / 32);

    const v4f* __restrict__ z4 = (const v4f*)z;
    v4f*       __restrict__ o4 = (v4f*)out;
    const v4f*              l4 = (const v4f*)ldsA;

    long r = gw;
    for (; r + 3 * nw < B_ROWS; r += 4 * nw) {
        const long r0 = r, r1 = r + nw, r2 = r + 2 * nw, r3 = r + 3 * nw;

        const int   c0 = idx[r0], c1 = idx[r1], c2 = idx[r2], c3 = idx[r3];
        const float s0 = scale[r0], s1 = scale[r1], s2 = scale[r2], s3 = scale[r3];

        // 4 independent NT stream-loads + 4 LDS gathers in flight
        v4f zv0 = __builtin_nontemporal_load(&z4[r0 * (ZDIM / 4) + lane]);
        v4f zv1 = __builtin_nontemporal_load(&z4[r1 * (ZDIM / 4) + lane]);
        v4f zv2 = __builtin_nontemporal_load(&z4[r2 * (ZDIM / 4) + lane]);
        v4f zv3 = __builtin_nontemporal_load(&z4[r3 * (ZDIM / 4) + lane]);

        v4f a0 = l4[(long)c0 * (ZDIM / 4) + lane];
        v4f a1 = l4[(long)c1 * (ZDIM / 4) + lane];
        v4f a2 = l4[(long)c2 * (ZDIM / 4) + lane];
        v4f a3 = l4[(long)c3 * (ZDIM / 4) + lane];

        v4f o0, o1, o2, o3;
        o0.x = fmaf(a0.x, s0, zv0.x); o0.y = fmaf(a0.y, s0, zv0.y);
        o0.z = fmaf(a0.z, s0, zv0.z); o0.w = fmaf(a0.w, s0, zv0.w);
        o1.x = fmaf(a1.x, s1, zv1.x); o1.y = fmaf(a1.y, s1, zv1.y);
        o1.z = fmaf(a1.z, s1, zv1.z); o1.w = fmaf(a1.w, s1, zv1.w);
        o2.x = fmaf(a2.x, s2, zv2.x); o2.y = fmaf(a2.y, s2, zv2.y);
        o2.z = fmaf(a2.z, s2, zv2.z); o2.w = fmaf(a2.w, s2, zv2.w);
        o3.x = fmaf(a3.x, s3, zv3.x); o3.y = fmaf(a3.y, s3, zv3.y);
        o3.z = fmaf(a3.z, s3, zv3.z); o3.w = fmaf(a3.w, s3, zv3.w);

        __builtin_nontemporal_store(o0, &o4[r0 * (ZDIM / 4) + lane]);
        __builtin_nontemporal_store(o1, &o4[r1 * (ZDIM / 4) + lane]);
        __builtin_nontemporal_store(o2, &o4[r2 * (ZDIM / 4) + lane]);
        __builtin_nontemporal_store(o3, &o4[r3 * (ZDIM / 4) + lane]);
    }
    for (; r < B_ROWS; r += nw) {   // tail (dead for this B/grid, kept for safety)
        const int   c = idx[r];
        const float s = scale[r];
        v4f zv = __builtin_nontemporal_load(&z4[r * (ZDIM / 4) + lane]);
        v4f av = l4[(long)c * (ZDIM / 4) + lane];
        v4f ov;
        ov.x = fmaf(av.x, s, zv.x);
        ov.y = fmaf(av.y, s, zv.y);
        ov.z = fmaf(av.z, s, zv.z);
        ov.w = fmaf(av.w, s, zv.w);
        __builtin_nontemporal_store(ov, &o4[r * (ZDIM / 4) + lane]);
    }
}

// ---------------------------------------------------------------
// Kernel 2: transpose a (128 x 512) -> aT (512 x 128), both sides coalesced.
// ---------------------------------------------------------------
__global__ __launch_bounds__(1024)
void A_trans_kernel(const float* __restrict__ a, float* __restrict__ aT) {
    __shared__ float tile[32][33];
    const int tx = threadIdx.x;          // 0..31
    const int ty = threadIdx.y;          // 0..31
    const int bx = blockIdx.x;           // 0..15  (c tiles)
    const int by = blockIdx.y;           // 0..3   (k tiles)

    const int c = bx * 32 + tx;
    const int k = by * 32 + ty;
    tile[ty][tx] = a[k * KDIM + c];      // coalesced read along c
    __syncthreads();

    const int c2 = bx * 32 + ty;
    const int k2 = by * 32 + tx;
    aT[c2 * ZDIM + k2] = tile[tx][ty];   // coalesced write along k
}

// ---------------------------------------------------------------
extern "C" void kernel_launch(void* const* d_in, const int* in_sizes, int n_in,
                              void* d_out, int out_size, void* d_ws, size_t ws_size,
                              hipStream_t stream) {
    const float* z     = (const float*)d_in[0];
    const float* a     = (const float*)d_in[1];
    const int*   idx   = (const int*)d_in[2];
    const float* scale = (const float*)d_in[3];
    float*       out   = (float*)d_out;
    float*       aT    = (float*)d_ws;    // 512*128*4 = 256 KB scratch

    A_trans_kernel<<<dim3(16, 4), dim3(32, 32), 0, stream>>>(a, aT);
    A_fused_kernel<<<NBLK, TBLK, 0, stream>>>(z, aT, idx, scale, out);
}